// MinimumSpanningTree_46978352283938
// MI455X (gfx1250) — compile-verified
//
#include <hip/hip_runtime.h>
#include <hip/hip_bf16.h>

// Problem constants (guide_in: [8, 256, 128, 128] f32)
#define Bb     8
#define Cc     256
#define Hh     128
#define Ww     128
#define Vv     (Hh * Ww)          // 16384 vertices
#define NVERT  ((Hh - 1) * Ww)    // 16256 vertical edges
#define NHORZ  (Hh * (Ww - 1))    // 16256 horizontal edges
#define Ee     (NVERT + NHORZ)    // 32512 edges per batch
#define GROUPS (Ee / 16)          // 2032 edge-groups of 16 per batch
#define NSORT  32768              // padded pow2 for bitonic sort

typedef __attribute__((ext_vector_type(2))) float v2f;
typedef __attribute__((ext_vector_type(8))) float v8f;

__device__ __forceinline__ void edge_uv(int e, int& u, int& v) {
    if (e < NVERT) {                       // vertical edge: (p, p+W)
        u = e; v = e + Ww;
    } else {                               // horizontal edge: (p, p+1)
        int t = e - NVERT;
        int r = t / (Ww - 1);
        int c = t - r * (Ww - 1);
        u = r * Ww + c; v = u + 1;
    }
}

// ---------------------------------------------------------------------------
// Kernel 1: per-edge squared-difference weights via V_WMMA_F32_16X16X4_F32.
// One wave handles 16 edges. Per K-chunk (4 channels), each lane supplies
// A[m][k] = diff(edge m, channel k); with the 16x4 f32 A-layout
// (lanes 0-15: M=0..15 / K={0,1}; lanes 16-31: M=0..15 / K={2,3}) the B=A^T
// operand has the identical per-lane contents, so a single v2f feeds both.
// diag(D) after 64 chunks = sum_c diff^2 — exact f32 products, f32 accum.
// ---------------------------------------------------------------------------
__global__ void mst_weights_wmma(const float* __restrict__ g,
                                 float* __restrict__ wout) {
    const int lane  = threadIdx.x & 31;
    const int wid   = blockIdx.x * (blockDim.x >> 5) + (threadIdx.x >> 5);
    const int b     = wid / GROUPS;
    const int grp   = wid - b * GROUPS;
    const int m     = lane & 15;       // edge slot within group (matrix row M)
    const int khalf = lane >> 4;       // which K pair this lane holds

    int u, v;
    edge_uv(grp * 16 + m, u, v);

    const float* gb = g + (size_t)b * Cc * Vv;

    v8f acc = {};
    #pragma unroll 4
    for (int ch = 0; ch < Cc / 4; ++ch) {
        const int c0 = ch * 4 + khalf * 2;
        const float* p0 = gb + (size_t)c0 * Vv;
        // pull the next channel block toward the caches while WMMA runs
        if (ch + 2 < Cc / 4) __builtin_prefetch(p0 + (size_t)8 * Vv + u, 0, 1);
        float d0 = p0[u]      - p0[v];
        float d1 = p0[Vv + u] - p0[Vv + v];
        v2f a; a.x = d0; a.y = d1;
        // D = A * A^T + C  (8 args: neg_a, A, neg_b, B, c_mod, C, reuse_a, reuse_b)
        acc = __builtin_amdgcn_wmma_f32_16x16x4_f32(
                  false, a, false, a, (short)0, acc, false, false);
    }

    // Diagonal extraction from the 16x16 f32 C/D layout:
    //   lanes 0-7  hold diag[M] at acc[M]          (M = lane)
    //   lanes 24-31 hold diag[M] at acc[M-8]       (M = lane-16)
    const bool valid = (lane < 8) || (lane >= 24);
    const int  r     = (lane < 8) ? lane : (lane - 24);
    float w = acc[0];
    if (r == 1) w = acc[1];
    if (r == 2) w = acc[2];
    if (r == 3) w = acc[3];
    if (r == 4) w = acc[4];
    if (r == 5) w = acc[5];
    if (r == 6) w = acc[6];
    if (r == 7) w = acc[7];

    if (valid) {
        const int eo = grp * 16 + ((lane < 8) ? lane : (lane - 16));
        wout[(size_t)b * Ee + eo] = w;
    }
}

// ---------------------------------------------------------------------------
// Kernel 2: per-batch stable sort of 32512 edges. Key = (f32 bits << 32) | idx
// (weights are sums of squares >= 0, so IEEE bits are order-isomorphic; the
// index in the low bits reproduces jnp.argsort's stable tie-breaking).
// Bitonic sort over 32768 u64 keys held entirely in 256 KB of LDS — only
// possible because CDNA5 allows up to 320 KB of LDS per workgroup.
// ---------------------------------------------------------------------------
__global__ void mst_sort(const float* __restrict__ w,
                         int* __restrict__ order) {
    extern __shared__ unsigned long long keys[];
    const int b   = blockIdx.x;
    const int tid = threadIdx.x;
    const float* wb = w + (size_t)b * Ee;

    for (int i = tid; i < NSORT; i += blockDim.x) {
        unsigned long long k;
        if (i < Ee) {
            unsigned int bits = __float_as_uint(wb[i]);
            k = ((unsigned long long)bits << 32) | (unsigned int)i;
        } else {
            k = ~0ull;   // pad sorts last
        }
        keys[i] = k;
    }
    __syncthreads();

    for (int kk = 2; kk <= NSORT; kk <<= 1) {
        for (int j = kk >> 1; j > 0; j >>= 1) {
            for (int i = tid; i < NSORT; i += blockDim.x) {
                const int ixj = i ^ j;
                if (ixj > i) {
                    unsigned long long a = keys[i];
                    unsigned long long c = keys[ixj];
                    const bool up = ((i & kk) == 0);
                    if ((a > c) == up) { keys[i] = c; keys[ixj] = a; }
                }
            }
            __syncthreads();
        }
    }

    for (int i = tid; i < Ee; i += blockDim.x)
        order[(size_t)b * Ee + i] = (int)(keys[i] & 0xffffffffu);
}

// ---------------------------------------------------------------------------
// Kernel 3: serial Kruskal per batch (inherently sequential), union-find state
// in LDS. Mirrors the reference exactly: path-halving find, union-by-rank with
// the same lo/hi selection, write index min(cnt, V-2), edges stored as (u,v).
// ---------------------------------------------------------------------------
__global__ void mst_kruskal(const int* __restrict__ order,
                            int* __restrict__ out) {
    extern __shared__ int lds[];
    int* parent = lds;        // Vv ints
    int* rnk    = lds + Vv;   // Vv ints
    const int b = blockIdx.x;

    for (int i = threadIdx.x; i < Vv; i += blockDim.x) {
        parent[i] = i;
        rnk[i]    = 0;
    }
    __syncthreads();

    if (threadIdx.x == 0) {
        const int* ob = order + (size_t)b * Ee;
        int* tb = out + (size_t)b * (Vv - 1) * 2;
        int cnt = 0;
        for (int s = 0; s < Ee; ++s) {
            int u, v;
            edge_uv(ob[s], u, v);
            // find(u) with path halving
            int ru = u;
            while (parent[ru] != ru) {
                int gp = parent[parent[ru]];
                parent[ru] = gp;
                ru = gp;
            }
            // find(v) with path halving
            int rv = v;
            while (parent[rv] != rv) {
                int gp = parent[parent[rv]];
                parent[rv] = gp;
                rv = gp;
            }
            if (ru != rv) {
                int lo, hi;
                if (rnk[ru] < rnk[rv]) { lo = ru; hi = rv; }
                else                   { lo = rv; hi = ru; }
                parent[lo] = hi;
                if (rnk[lo] == rnk[hi]) rnk[hi] += 1;
                const int idx = (cnt < (Vv - 2)) ? cnt : (Vv - 2);
                tb[idx * 2 + 0] = u;
                tb[idx * 2 + 1] = v;
                ++cnt;
            }
        }
    }
}

extern "C" void kernel_launch(void* const* d_in, const int* in_sizes, int n_in,
                              void* d_out, int out_size, void* d_ws, size_t ws_size,
                              hipStream_t stream) {
    (void)in_sizes; (void)n_in; (void)out_size; (void)ws_size;

    const float* guide = (const float*)d_in[0];

    // Workspace layout: [ f32 weights B*E | int sorted-edge-ids B*E ]
    float* wts = (float*)d_ws;
    size_t off = ((size_t)Bb * Ee * sizeof(float) + 255) & ~(size_t)255;
    int* order = (int*)((char*)d_ws + off);

    // 1) weights: 8 waves/block, one wave per 16-edge group, B*GROUPS waves total
    mst_weights_wmma<<<GROUPS, 256, 0, stream>>>(guide, wts);

    // 2) per-batch stable bitonic sort in 256 KB LDS
    mst_sort<<<Bb, 1024, NSORT * sizeof(unsigned long long), stream>>>(wts, order);

    // 3) per-batch serial Kruskal, union-find in 128 KB LDS
    mst_kruskal<<<Bb, 256, 2 * Vv * sizeof(int), stream>>>(order, (int*)d_out);
}